// GraphLayer_12206297055245
// MI455X (gfx1250) — compile-verified
//
#include <hip/hip_runtime.h>

#define D_FEAT 64
#define EPS_LN 1e-5f

typedef __attribute__((ext_vector_type(2))) float v2f;
typedef __attribute__((ext_vector_type(8))) float v8f;

// ---------------- degree / norm ----------------
__global__ void k_deg_init(float* __restrict__ deg, int n) {
    int i = blockIdx.x * blockDim.x + threadIdx.x;
    if (i < n) deg[i] = 1.0f;  // self loop
}

__global__ void k_deg_count(const int* __restrict__ dst, float* __restrict__ deg, int e) {
    int i = blockIdx.x * blockDim.x + threadIdx.x;
    if (i < e) atomicAdd(&deg[dst[i]], 1.0f);
}

__global__ void k_deg_rsqrt(float* __restrict__ deg, int n) {
    int i = blockIdx.x * blockDim.x + threadIdx.x;
    if (i < n) deg[i] = rsqrtf(deg[i]);
}

// ---------------- dense projection h = X @ W via V_WMMA_F32_16X16X4_F32 ----
// W is pre-swizzled into LDS in per-lane fragment order so that each lane's
// (b.x, b.y) pair for (k-step ks, n-tile nt) is a single contiguous b64:
//   ldsB[((ks*4 + nt)*32 + lane)*2 + {0,1}]
// A (16x4 f32): lane<16 -> row=lane, K={k0,k0+1}; lane>=16 -> row=lane-16, K={k0+2,k0+3}
// C/D (16x16 f32): 8 VGPRs, lane half selects M / M+8.
template <int MT>
__device__ __forceinline__ void gemm_rows(const float* __restrict__ X,
                                          const float* __restrict__ ldsB,
                                          float* __restrict__ H, int m0, int lane) {
    int rowoff = lane & 15;
    int khalf  = (lane >> 4) << 1;  // 0 or 2
    int col    = lane & 15;

    v8f acc[MT][4];
#pragma unroll
    for (int t = 0; t < MT; ++t)
#pragma unroll
        for (int j = 0; j < 4; ++j) acc[t][j] = (v8f){};

#pragma unroll
    for (int ks = 0; ks < 16; ++ks) {
        v2f a[MT];
#pragma unroll
        for (int t = 0; t < MT; ++t) {
            const float* xp = X + (m0 + t * 16 + rowoff) * D_FEAT + ks * 4 + khalf;
            a[t].x = xp[0];
            a[t].y = xp[1];
        }
#pragma unroll
        for (int j = 0; j < 4; ++j) {
            v2f b = *(const v2f*)(ldsB + (((ks * 4 + j) * 32) + lane) * 2);  // ds_load_b64
#pragma unroll
            for (int t = 0; t < MT; ++t)
                acc[t][j] = __builtin_amdgcn_wmma_f32_16x16x4_f32(
                    false, a[t], false, b, (short)0, acc[t][j], false, false);
        }
    }

    int rbase = (lane >> 4) << 3;  // 0 or 8
#pragma unroll
    for (int t = 0; t < MT; ++t)
#pragma unroll
        for (int v = 0; v < 8; ++v) {
            float* hp = H + (m0 + t * 16 + rbase + v) * D_FEAT + col;
            hp[0]  = acc[t][0][v];
            hp[16] = acc[t][1][v];
            hp[32] = acc[t][2][v];
            hp[48] = acc[t][3][v];
        }
}

__global__ __launch_bounds__(256) void k_gemm_wmma(const float* __restrict__ X,
                                                   const float* __restrict__ W,
                                                   float* __restrict__ H, int n) {
    __shared__ float ldsB[D_FEAT * D_FEAT];  // 16 KB, fragment-swizzled
    int tid = threadIdx.x;
    // swizzle fill: pair p -> (k-step, n-tile, lane)
#pragma unroll
    for (int p = tid; p < 2048; p += 256) {
        int ks  = p >> 7;        // 0..15
        int rem = p & 127;
        int nt  = rem >> 5;      // 0..3
        int ln  = rem & 31;
        int kh  = (ln >> 4) << 1;
        int cl  = ln & 15;
        int k   = ks * 4 + kh;
        ldsB[p * 2]     = W[k * D_FEAT + nt * 16 + cl];
        ldsB[p * 2 + 1] = W[(k + 1) * D_FEAT + nt * 16 + cl];
    }
    __syncthreads();

    int wave   = tid >> 5;
    int lane   = tid & 31;
    int mtiles = n >> 4;                 // 3125 for n=50000
    int wid    = blockIdx.x * 8 + wave;  // one wave handles 2 m-tiles
    int mt0    = wid * 2;
    if (mt0 >= mtiles) return;           // wave-uniform -> EXEC all-ones inside
    if (mt0 + 1 < mtiles)
        gemm_rows<2>(X, ldsB, H, mt0 << 4, lane);
    else
        gemm_rows<1>(X, ldsB, H, mt0 << 4, lane);
}

// ---------------- self-loop + bias (plain write -> deterministic init) -----
__global__ void k_selfloop_bias(const float* __restrict__ h, const float* __restrict__ dinv,
                                const float* __restrict__ bias, float* __restrict__ out, int total) {
    int idx = blockIdx.x * blockDim.x + threadIdx.x;
    if (idx < total) {
        int node = idx >> 6;
        int j    = idx & 63;
        float di = dinv[node];
        out[idx] = bias[j] + di * di * h[idx];
    }
}

// ---------------- edge scatter: 16 threads per edge, float4 per thread -----
__global__ void k_edge_scatter(const int* __restrict__ src, const int* __restrict__ dst,
                               const float* __restrict__ h, const float* __restrict__ dinv,
                               float* __restrict__ out, int e) {
    int t    = blockIdx.x * blockDim.x + threadIdx.x;
    int edge = t >> 4;
    if (edge >= e) return;
    int seg = (t & 15) << 2;  // feature offset 0,4,...,60
    int s = src[edge];
    int d = dst[edge];
    float norm = dinv[s] * dinv[d];
    float4 m = *(const float4*)(h + s * D_FEAT + seg);  // L2-resident gather
    float* op = out + d * D_FEAT + seg;
    atomicAdd(op + 0, m.x * norm);
    atomicAdd(op + 1, m.y * norm);
    atomicAdd(op + 2, m.z * norm);
    atomicAdd(op + 3, m.w * norm);
}

// ---------------- PReLU (shared scalar a) ----------------------------------
__global__ void k_prelu(float* __restrict__ x, const float* __restrict__ a_ptr, int total) {
    int i = blockIdx.x * blockDim.x + threadIdx.x;
    if (i < total) {
        float a = *a_ptr;
        float v = x[i];
        x[i] = v >= 0.0f ? v : a * v;
    }
}

// ---------------- LayerNorm: one wave32 per node, 2 feats per lane ---------
__global__ __launch_bounds__(256) void k_layernorm(float* __restrict__ x,
                                                   const float* __restrict__ gamma,
                                                   const float* __restrict__ beta, int n) {
    int wave = threadIdx.x >> 5;
    int lane = threadIdx.x & 31;
    int node = blockIdx.x * 8 + wave;
    if (node >= n) return;
    float* xp = x + node * D_FEAT;
    float v0 = xp[lane];
    float v1 = xp[lane + 32];
    float s = v0 + v1;
#pragma unroll
    for (int off = 16; off > 0; off >>= 1) s += __shfl_xor(s, off, 32);
    float mu = s * (1.0f / 64.0f);
    float d0 = v0 - mu, d1 = v1 - mu;
    float q = d0 * d0 + d1 * d1;
#pragma unroll
    for (int off = 16; off > 0; off >>= 1) q += __shfl_xor(q, off, 32);
    float inv = rsqrtf(q * (1.0f / 64.0f) + EPS_LN);
    xp[lane]      = d0 * inv * gamma[lane]      + beta[lane];
    xp[lane + 32] = d1 * inv * gamma[lane + 32] + beta[lane + 32];
}

extern "C" void kernel_launch(void* const* d_in, const int* in_sizes, int n_in,
                              void* d_out, int out_size, void* d_ws, size_t ws_size,
                              hipStream_t stream) {
    const float* X     = (const float*)d_in[0];
    const float* W1    = (const float*)d_in[1];
    const float* b1    = (const float*)d_in[2];
    const float* W2    = (const float*)d_in[3];
    const float* b2    = (const float*)d_in[4];
    const float* W3    = (const float*)d_in[5];
    const float* b3    = (const float*)d_in[6];
    const float* pa    = (const float*)d_in[7];
    const float* gamma = (const float*)d_in[8];
    const float* beta  = (const float*)d_in[9];
    const int*   ei    = (const int*)d_in[10];

    int n = in_sizes[0] / D_FEAT;  // 50000
    int e = in_sizes[10] / 2;      // 1250000
    const int* src = ei;
    const int* dst = ei + e;

    float* out  = (float*)d_out;
    float* dinv = (float*)d_ws;  // n floats
    float* bufH = dinv + n;      // n*64 floats (GEMM output, L2-resident)

    int total  = n * D_FEAT;
    int mtiles = n >> 4;
    int bN     = (n + 255) / 256;
    int bE     = (e + 255) / 256;
    int bT     = (total + 255) / 256;
    int bScat  = (int)(((long long)e * 16 + 255) / 256);
    int bGemm  = (((mtiles + 1) / 2) + 7) / 8;
    int bLN    = (n + 7) / 8;

    // shared symmetric normalization (computed once, reused by all layers)
    k_deg_init <<<bN, 256, 0, stream>>>(dinv, n);
    k_deg_count<<<bE, 256, 0, stream>>>(dst, dinv, e);
    k_deg_rsqrt<<<bN, 256, 0, stream>>>(dinv, n);

    // ---- layer 1 ----
    k_gemm_wmma    <<<bGemm, 256, 0, stream>>>(X, W1, bufH, n);
    k_selfloop_bias<<<bT,    256, 0, stream>>>(bufH, dinv, b1, out, total);
    k_edge_scatter <<<bScat, 256, 0, stream>>>(src, dst, bufH, dinv, out, e);
    k_prelu        <<<bT,    256, 0, stream>>>(out, pa, total);

    // ---- layer 2 ----
    k_gemm_wmma    <<<bGemm, 256, 0, stream>>>(out, W2, bufH, n);
    k_selfloop_bias<<<bT,    256, 0, stream>>>(bufH, dinv, b2, out, total);
    k_edge_scatter <<<bScat, 256, 0, stream>>>(src, dst, bufH, dinv, out, e);
    k_prelu        <<<bT,    256, 0, stream>>>(out, pa, total);

    // ---- layer 3 + LayerNorm ----
    k_gemm_wmma    <<<bGemm, 256, 0, stream>>>(out, W3, bufH, n);
    k_selfloop_bias<<<bT,    256, 0, stream>>>(bufH, dinv, b3, out, total);
    k_edge_scatter <<<bScat, 256, 0, stream>>>(src, dst, bufH, dinv, out, e);
    k_layernorm    <<<bLN,   256, 0, stream>>>(out, gamma, beta, n);
}